// TADPooling_66975720014378
// MI455X (gfx1250) — compile-verified
//
#include <hip/hip_runtime.h>
#include <math.h>

#define NN 2048
#define CD 256
#define C2 512
#define EE 131072
#define LEO 8
#define FD 16
#define SLOPEV 0.2f

// GEMM tiling
#define BM 64
#define BN 64
#define BKK 16
#define ANPAD 20   // padded K-stride for non-transposed A tiles in LDS

typedef float v2f __attribute__((ext_vector_type(2)));
typedef float v8f __attribute__((ext_vector_type(8)));
typedef int   v4i __attribute__((vector_size(16)));

#if defined(__has_builtin)
#  if __has_builtin(__builtin_amdgcn_wmma_f32_16x16x4_f32)
#    define HAVE_WMMA 1
#  endif
#  if __has_builtin(__builtin_amdgcn_global_load_async_to_lds_b128)
#    define HAVE_ASYNC 1
#  endif
#  if __has_builtin(__builtin_amdgcn_s_wait_asynccnt)
#    define HAVE_WAIT_ASYNC_BUILTIN 1
#  endif
#endif
#ifndef HAVE_WMMA
#  define HAVE_WMMA 0
#endif
#ifndef HAVE_ASYNC
#  define HAVE_ASYNC 0
#endif
#ifndef HAVE_WAIT_ASYNC_BUILTIN
#  define HAVE_WAIT_ASYNC_BUILTIN 0
#endif

// ---------------- device helpers ----------------
__device__ __forceinline__ void atomicMaxF(float* addr, float v) {
  if (v >= 0.0f) atomicMax((int*)addr, __float_as_int(v));
  else           atomicMin((unsigned int*)addr, (unsigned int)__float_as_int(v));
}
__device__ __forceinline__ float leakyf(float s) { return s >= 0.0f ? s : SLOPEV * s; }
__device__ __forceinline__ float sigmoidf_(float v) { return 1.0f / (1.0f + __expf(-v)); }

// 16-byte global -> LDS copy; async DMA (ASYNCcnt) when the builtin exists.
__device__ __forceinline__ void copy16_g2l(const float* gsrc, float* ldst) {
#if HAVE_ASYNC
  __builtin_amdgcn_global_load_async_to_lds_b128(
      (__attribute__((address_space(1))) v4i*)(v4i*)const_cast<float*>(gsrc),
      (__attribute__((address_space(3))) v4i*)(v4i*)ldst,
      0, 0);
#else
  *(float4*)ldst = *(const float4*)gsrc;
#endif
}

__device__ __forceinline__ void wait_async_loads() {
#if HAVE_ASYNC
#  if HAVE_WAIT_ASYNC_BUILTIN
  __builtin_amdgcn_s_wait_asynccnt(0);
#  else
  asm volatile("s_wait_asynccnt 0x0" ::: "memory");
#  endif
#endif
}

// ---------------- fills ----------------
__global__ void fill_f32(float* __restrict__ p, float v, int n) {
  int i = blockIdx.x * blockDim.x + threadIdx.x;
  if (i < n) p[i] = v;
}

// ---------------- LDS-tiled WMMA GEMM ----------------
// D[M,N] = op(A)[M,K] @ B[K,N] (+bias) * rowscale * colscale, optional diag zero.
// 256 threads = 8 wave32s; block tile 64x64; wave tile 16x32 (2 accumulators).
// K-panels of depth 16 staged in LDS, double-buffered, loaded with
// GLOBAL_LOAD_ASYNC_TO_LDS_B128 (ASYNCcnt) when available.
__global__ __launch_bounds__(256)
void wmma_gemm_f32(const float* __restrict__ A, int lda, int transA,
                   const float* __restrict__ B, int ldb, int K,
                   const float* __restrict__ bias,
                   const float* __restrict__ rowscale,
                   const float* __restrict__ colscale,
                   int zero_diag,
                   float* __restrict__ D, int ldd) {
  // As: trans layout  -> [k][m], BKK*BM floats (contiguous in m)
  //     non-trans     -> [m][ANPAD], BM*ANPAD floats (16B-aligned rows,
  //                      conflict-free stride-20 fragment reads)
  __shared__ __align__(16) float As[2][BM * ANPAD];
  __shared__ __align__(16) float Bs[2][BKK * BN];

  const int tid   = threadIdx.x;
  const int lane  = tid & 31;
  const int wid   = tid >> 5;       // 0..7
  const int wm    = wid & 3;        // 4 waves along M
  const int wn    = wid >> 2;       // 2 waves along N
  const int n0    = blockIdx.x * BN;
  const int m0    = blockIdx.y * BM;
  const int mfrag = lane & 15;
  const int kfrag = (lane >> 4) * 2;

  v8f acc0 = {0.f, 0.f, 0.f, 0.f, 0.f, 0.f, 0.f, 0.f};
  v8f acc1 = {0.f, 0.f, 0.f, 0.f, 0.f, 0.f, 0.f, 0.f};

  const int nPanels = K / BKK;

  // one B128 (4 floats) per thread per matrix per panel
  auto load_panel = [&](int buf, int k0) {
    const int p = tid * 4;
    // ---- A panel (BM x BKK = 1024 floats) ----
    const float* gsrcA;
    float* ldstA;
    if (transA) {                       // A'(m,k) = A[k*lda + m]; contiguous in m
      int ka = p >> 6, ma = p & 63;
      gsrcA = A + (size_t)(k0 + ka) * lda + (m0 + ma);
      ldstA = &As[buf][ka * BM + ma];
    } else {                            // contiguous in k
      int ma = p >> 4, ka = p & 15;
      gsrcA = A + (size_t)(m0 + ma) * lda + (k0 + ka);
      ldstA = &As[buf][ma * ANPAD + ka];
    }
    // ---- B panel (BKK x BN = 1024 floats); contiguous in n ----
    int kb = p >> 6, nb = p & 63;
    const float* gsrcB = B + (size_t)(k0 + kb) * ldb + (n0 + nb);
    float* ldstB = &Bs[buf][kb * BN + nb];
    copy16_g2l(gsrcA, ldstA);
    copy16_g2l(gsrcB, ldstB);
  };

  load_panel(0, 0);
  for (int t = 0; t < nPanels; ++t) {
    const int buf = t & 1;
    wait_async_loads();     // this wave's panel-t loads are in LDS
    __syncthreads();        // every wave's panel-t data visible; panel t-1 reads done
    if (t + 1 < nPanels) load_panel(buf ^ 1, (t + 1) * BKK);

    const int mL = wm * 16 + mfrag;
    const int nL = wn * 32 + mfrag;
#if HAVE_WMMA
    for (int kk = 0; kk < BKK; kk += 4) {
      v2f av, b0, b1;
      if (transA) {
        av.x = As[buf][(kk + kfrag)     * BM + mL];
        av.y = As[buf][(kk + kfrag + 1) * BM + mL];
      } else {
        av.x = As[buf][mL * ANPAD + (kk + kfrag)];
        av.y = As[buf][mL * ANPAD + (kk + kfrag + 1)];
      }
      b0.x = Bs[buf][(kk + kfrag)     * BN + nL];
      b0.y = Bs[buf][(kk + kfrag + 1) * BN + nL];
      b1.x = Bs[buf][(kk + kfrag)     * BN + nL + 16];
      b1.y = Bs[buf][(kk + kfrag + 1) * BN + nL + 16];
      acc0 = __builtin_amdgcn_wmma_f32_16x16x4_f32(false, av, false, b0, (short)0, acc0,
                                                   false, false);
      acc1 = __builtin_amdgcn_wmma_f32_16x16x4_f32(false, av, false, b1, (short)0, acc1,
                                                   false, false);
    }
#else
    // scalar fallback with identical C/D fragment ownership
    {
      const int rowL = wm * 16 + ((lane >> 4) ? 8 : 0);
      const int colL = wn * 32 + (lane & 15);
      for (int r = 0; r < 8; ++r) {
        float s0 = acc0[r], s1 = acc1[r];
        for (int kk2 = 0; kk2 < BKK; ++kk2) {
          float a = transA ? As[buf][kk2 * BM + rowL + r]
                           : As[buf][(rowL + r) * ANPAD + kk2];
          s0 += a * Bs[buf][kk2 * BN + colL];
          s1 += a * Bs[buf][kk2 * BN + colL + 16];
        }
        acc0[r] = s0; acc1[r] = s1;
      }
    }
#endif
  }

  // epilogue: wave owns rows m0+wm*16..+15, cols n0+wn*32..+31
  const int rbase = m0 + wm * 16 + ((lane >> 4) ? 8 : 0);
  for (int t2 = 0; t2 < 2; ++t2) {
    const int col = n0 + wn * 32 + t2 * 16 + (lane & 15);
    const float cs = colscale ? colscale[col] : 1.0f;
    const float bb = bias ? bias[col] : 0.0f;
    for (int r = 0; r < 8; ++r) {
      const int row = rbase + r;
      float v = (t2 ? acc1[r] : acc0[r]) + bb;
      if (rowscale) v *= rowscale[row];
      v *= cs;
      if (zero_diag && row == col) v = 0.0f;
      D[(size_t)row * ldd + col] = v;
    }
  }
}

// ---------------- graph stages ----------------
__global__ void chain_xq_kernel(const float* __restrict__ x, float* __restrict__ xq_s) {
  int idx = blockIdx.x * blockDim.x + threadIdx.x;
  if (idx >= NN * CD) return;
  int i = idx / CD, c = idx % CD;
  float v = -INFINITY;
  if (i > 0)      v = x[(i - 1) * CD + c];
  if (i < NN - 1) v = fmaxf(v, x[(i + 1) * CD + c]);
  xq_s[idx] = v;
}

__global__ void scatter_max_kernel(const int* __restrict__ ei, const float* __restrict__ x,
                                   float* __restrict__ xq_t) {
  int idx = blockIdx.x * blockDim.x + threadIdx.x;   // EE*CD = 33.5M < 2^31
  if (idx >= EE * CD) return;
  int e = idx / CD, c = idx % CD;
  int src = ei[e], dst = ei[EE + e];
  atomicMaxF(&xq_t[dst * CD + c], x[src * CD + c]);
}

__global__ void fix_neginf(float* __restrict__ p, int n) {
  int i = blockIdx.x * blockDim.x + threadIdx.x;
  if (i < n && p[i] == -INFINITY) p[i] = 0.0f;
}

__global__ void node_dots_kernel(const float* __restrict__ x,
                                 const float* __restrict__ xqlin_t,
                                 const float* __restrict__ xqlin_s,
                                 const float* __restrict__ att_t_w,
                                 const float* __restrict__ att_s_w,
                                 float* __restrict__ aq_t, float* __restrict__ aj_t,
                                 float* __restrict__ aq_s, float* __restrict__ aj_s) {
  int idx = blockIdx.x * blockDim.x + threadIdx.x;
  if (idx >= NN * 4) return;
  int i = idx >> 2, which = idx & 3;
  const float* vec; const float* wv; float* outp;
  switch (which) {
    case 0:  vec = &xqlin_t[i * CD]; wv = att_t_w;      outp = &aq_t[i]; break;
    case 1:  vec = &x[i * CD];       wv = att_t_w + CD; outp = &aj_t[i]; break;
    case 2:  vec = &xqlin_s[i * CD]; wv = att_s_w;      outp = &aq_s[i]; break;
    default: vec = &x[i * CD];       wv = att_s_w + CD; outp = &aj_s[i]; break;
  }
  float s = 0.0f;
  for (int k = 0; k < CD; ++k) s += vec[k] * wv[k];
  *outp = s;
}

__global__ void edge_scoreA(const int* __restrict__ ei, const float* __restrict__ aq,
                            const float* __restrict__ aj, const float* __restrict__ attb,
                            float* __restrict__ scores, float* __restrict__ smax,
                            int* __restrict__ indeg) {
  int e = blockIdx.x * blockDim.x + threadIdx.x;
  if (e >= EE) return;
  int src = ei[e], dst = ei[EE + e];
  float s = leakyf(aq[dst] + aj[src] + attb[0]);
  scores[e] = s;
  atomicMaxF(&smax[dst], s);
  atomicAdd(&indeg[dst], 1);
}

__global__ void edge_scoreB(const int* __restrict__ ei, float* __restrict__ scores,
                            const float* __restrict__ smax, float* __restrict__ ssum) {
  int e = blockIdx.x * blockDim.x + threadIdx.x;
  if (e >= EE) return;
  int dst = ei[EE + e];
  float v = __expf(scores[e] - smax[dst]);
  scores[e] = v;
  atomicAdd(&ssum[dst], v);
}

__global__ __launch_bounds__(256)
void edge_scatter_xt(const int* __restrict__ ei, const float* __restrict__ x,
                     float* __restrict__ scores, const float* __restrict__ ssum,
                     float* __restrict__ xc) {
  int e = blockIdx.x;
  int c = threadIdx.x;
  int src = ei[e], dst = ei[EE + e];
  float raw = scores[e];
  float sc  = raw / (ssum[dst] + 1e-16f);
  __syncthreads();
  if (c == 0) scores[e] = sc;   // final normalized score (used for dense S later)
  atomicAdd(&xc[dst * C2 + c], sc * x[src * CD + c]);
}

__global__ __launch_bounds__(256)
void chain_xs(const float* __restrict__ x, const float* __restrict__ aq_s,
              const float* __restrict__ aj_s, const float* __restrict__ attb,
              float* __restrict__ xc) {
  int i = blockIdx.x;
  int c = threadIdx.x;
  float aq = aq_s[i], b = attb[0];
  bool hl = i > 0, hr = i < NN - 1;
  float sl = hl ? leakyf(aq + aj_s[i - 1] + b) : -INFINITY;
  float sr = hr ? leakyf(aq + aj_s[i + 1] + b) : -INFINITY;
  float m  = fmaxf(sl, sr);
  float el = hl ? __expf(sl - m) : 0.0f;
  float er = hr ? __expf(sr - m) : 0.0f;
  float inv = 1.0f / (el + er + 1e-16f);
  float v = 0.0f;
  if (hl) v += (el * inv) * x[(i - 1) * CD + c];
  if (hr) v += (er * inv) * x[(i + 1) * CD + c];
  xc[i * C2 + CD + c] = v;
}

__global__ __launch_bounds__(64)
void node_h_kernel(const float* __restrict__ xc,
                   const float* w1t, const float* b1t, const float* w2t,
                   const float* w3t, const float* b3t,
                   const float* w1s, const float* b1s, const float* w2s,
                   const float* w3s, const float* b3s,
                   const int* __restrict__ indeg_t,
                   float* __restrict__ h1_t, float* __restrict__ h1_s,
                   float* __restrict__ fitbase_t, float* __restrict__ fitbase_s) {
  int i = blockIdx.x;
  int tid = threadIdx.x;
  __shared__ float sh[6][LEO];
  if (tid < 48) {
    int which = tid >> 3, o = tid & 7;
    const float* w; float bias = 0.0f;
    switch (which) {
      case 0:  w = w1t; bias = b1t[o]; break;
      case 1:  w = w2t; break;
      case 2:  w = w3t; bias = b3t[o]; break;
      case 3:  w = w1s; bias = b1s[o]; break;
      case 4:  w = w2s; break;
      default: w = w3s; bias = b3s[o]; break;
    }
    float s = bias;
    const float* row = &xc[i * C2];
    for (int k = 0; k < C2; ++k) s += row[k] * w[k * LEO + o];
    sh[which][o] = s;
  }
  __syncthreads();
  if (tid < LEO) {
    h1_t[i * LEO + tid]      = sh[0][tid];
    fitbase_t[i * LEO + tid] = sh[2][tid] - (float)indeg_t[i] * sh[1][tid];
    h1_s[i * LEO + tid]      = sh[3][tid];
    float degs = (float)((i > 0) + (i < NN - 1));
    fitbase_s[i * LEO + tid] = sh[5][tid] - degs * sh[4][tid];
  }
}

__global__ void edge_scatter_fit(const int* __restrict__ ei, const float* __restrict__ h1_t,
                                 float* __restrict__ fitacc) {
  int idx = blockIdx.x * blockDim.x + threadIdx.x;  // EE*8
  if (idx >= EE * LEO) return;
  int e = idx >> 3, o = idx & 7;
  int src = ei[e], dst = ei[EE + e];
  atomicAdd(&fitacc[dst * LEO + o], h1_t[src * LEO + o]);
}

__global__ void fit_assemble(const float* __restrict__ fitacc, const float* __restrict__ fitbase_t,
                             const float* __restrict__ fitbase_s, const float* __restrict__ h1_s,
                             float* __restrict__ fit) {
  int idx = blockIdx.x * blockDim.x + threadIdx.x;  // NN*16
  if (idx >= NN * FD) return;
  int i = idx >> 4, col = idx & 15;
  float v;
  if (col < LEO) {
    v = fitacc[i * LEO + col] + fitbase_t[i * LEO + col];
  } else {
    int o = col - LEO;
    v = fitbase_s[i * LEO + o];
    if (i > 0)      v += h1_s[(i - 1) * LEO + o];
    if (i < NN - 1) v += h1_s[(i + 1) * LEO + o];
  }
  fit[idx] = v;
}

__global__ __launch_bounds__(256)
void se_gate_kernel(const float* __restrict__ fit, const float* __restrict__ w1,
                    const float* __restrict__ b1, const float* __restrict__ w2,
                    const float* __restrict__ b2, float* __restrict__ gate) {
  __shared__ float sh[256];
  __shared__ float hid[4];
  int tid = threadIdx.x;
  int col = tid & 15, part = tid >> 4;
  float s = 0.0f;
  for (int r = part; r < NN; r += 16) s += fit[r * FD + col];
  sh[tid] = s;
  __syncthreads();
  for (int st = 8; st > 0; st >>= 1) {
    if (part < st) sh[part * 16 + col] += sh[(part + st) * 16 + col];
    __syncthreads();
  }
  if (tid < 4) {
    float a = b1[tid];
    for (int k = 0; k < FD; ++k) a += (sh[k] / (float)NN) * w1[k * 4 + tid];
    hid[tid] = fmaxf(a, 0.0f);
  }
  __syncthreads();
  if (tid < FD) {
    float g = b2[tid];
    for (int k = 0; k < 4; ++k) g += hid[k] * w2[k * FD + tid];
    gate[tid] = sigmoidf_(g);
  }
}

__global__ void fr_kernel(const float* __restrict__ fit, const float* __restrict__ gate,
                          float* __restrict__ fr) {
  int i = blockIdx.x * blockDim.x + threadIdx.x;
  if (i >= NN) return;
  float s = 0.0f;
  for (int o = 0; o < FD; ++o) s += fit[i * FD + o] * gate[o];
  fr[i] = s;
}

__global__ void conv_kernel(const float* __restrict__ fr, const float* __restrict__ cw,
                            const float* __restrict__ cb, float* __restrict__ fout) {
  int i = blockIdx.x * blockDim.x + threadIdx.x;
  if (i >= NN) return;
  float a = (i > 0)      ? fr[i - 1] : 0.0f;
  float c = (i < NN - 1) ? fr[i + 1] : 0.0f;
  fout[i] = sigmoidf_(cw[0] * a + cw[1] * fr[i] + cw[2] * c + cb[0]);
}

__global__ void window_scatter(const float* __restrict__ f, int* __restrict__ wm) {
  int i = blockIdx.x * blockDim.x + threadIdx.x;
  if (i >= NN) return;
  float best = -INFINITY; int bestj = max(i - 15, 0);
  for (int d = -15; d <= 15; ++d) {
    int j = i + d;
    bool valid = (j >= 0) && (j < NN);
    int jc = min(max(j, 0), NN - 1);
    float v = valid ? f[jc] : -INFINITY;
    if (v > best) { best = v; bestj = jc; }
  }
  atomicOr(&wm[bestj], 1);
}

__global__ void mask_combine(const float* __restrict__ f, const int* __restrict__ wm,
                             int* __restrict__ maski, int* __restrict__ flags) {
  int i = blockIdx.x * blockDim.x + threadIdx.x;
  if (i >= NN) return;
  bool peak = (i > 0) && (i < NN - 1) && (f[i - 1] < f[i]) && (f[i + 1] < f[i]);
  int m = (wm[i] && peak) ? 1 : 0;
  maski[i] = m;
  if (m) atomicOr(&flags[0], 1);
}

__global__ __launch_bounds__(256)
void argmax_kernel(const float* __restrict__ f, int* __restrict__ flags) {
  __shared__ float sv[256]; __shared__ int si[256];
  int tid = threadIdx.x;
  float best = -INFINITY; int bi = 0;
  for (int i = tid; i < NN; i += 256) { float v = f[i]; if (v > best) { best = v; bi = i; } }
  sv[tid] = best; si[tid] = bi;
  __syncthreads();
  for (int st = 128; st > 0; st >>= 1) {
    if (tid < st) {
      if (sv[tid + st] > sv[tid] || (sv[tid + st] == sv[tid] && si[tid + st] < si[tid])) {
        sv[tid] = sv[tid + st]; si[tid] = si[tid + st];
      }
    }
    __syncthreads();
  }
  if (tid == 0) flags[1] = si[0];
}

__global__ void mask_final(const int* __restrict__ maski, const int* __restrict__ flags,
                           float* __restrict__ maskf, float* __restrict__ mask_out) {
  int i = blockIdx.x * blockDim.x + threadIdx.x;
  if (i >= NN) return;
  int m = flags[0] ? maski[i] : (i == flags[1] ? 1 : 0);
  maskf[i] = (float)m;
  mask_out[i] = (float)m;
}

__global__ void xout_kernel(const float* __restrict__ xc, const float* __restrict__ f,
                            const float* __restrict__ maskf, float* __restrict__ xo) {
  int idx = blockIdx.x * blockDim.x + threadIdx.x;  // NN*C2
  if (idx >= NN * C2) return;
  int i = idx / C2;
  xo[idx] = xc[idx] * f[i] * maskf[i];
}

__global__ void build_AS(const int* __restrict__ ei, const float* __restrict__ ew,
                         const float* __restrict__ scores,
                         float* __restrict__ Ad, float* __restrict__ Sd) {
  int e = blockIdx.x * blockDim.x + threadIdx.x;
  if (e >= EE) return;
  int src = ei[e], dst = ei[EE + e];
  atomicAdd(&Ad[(size_t)src * NN + dst], ew[e]);
  atomicAdd(&Sd[(size_t)src * NN + dst], scores[e]);
}

// ---------------- host ----------------
extern "C" void kernel_launch(void* const* d_in, const int* in_sizes, int n_in,
                              void* d_out, int out_size, void* d_ws, size_t ws_size,
                              hipStream_t stream) {
  const float* x        = (const float*)d_in[0];
  const int*   ei       = (const int*)  d_in[1];
  const float* ew       = (const float*)d_in[2];
  const float* lin_t_w  = (const float*)d_in[3];
  const float* lin_t_b  = (const float*)d_in[4];
  const float* att_t_w  = (const float*)d_in[5];
  const float* att_t_b  = (const float*)d_in[6];
  const float* lin_s_w  = (const float*)d_in[7];
  const float* lin_s_b  = (const float*)d_in[8];
  const float* att_s_w  = (const float*)d_in[9];
  const float* att_s_b  = (const float*)d_in[10];
  const float* le_t_w1  = (const float*)d_in[11];
  const float* le_t_b1  = (const float*)d_in[12];
  const float* le_t_w2  = (const float*)d_in[13];
  const float* le_t_w3  = (const float*)d_in[14];
  const float* le_t_b3  = (const float*)d_in[15];
  const float* le_s_w1  = (const float*)d_in[16];
  const float* le_s_b1  = (const float*)d_in[17];
  const float* le_s_w2  = (const float*)d_in[18];
  const float* le_s_w3  = (const float*)d_in[19];
  const float* le_s_b3  = (const float*)d_in[20];
  const float* se_w1    = (const float*)d_in[21];
  const float* se_b1    = (const float*)d_in[22];
  const float* se_w2    = (const float*)d_in[23];
  const float* se_b2    = (const float*)d_in[24];
  const float* conv_w   = (const float*)d_in[25];
  const float* conv_b   = (const float*)d_in[26];

  float* out      = (float*)d_out;
  float* xout     = out;                              // [NN, C2]
  float* fout     = xout + (size_t)NN * C2;           // [NN]
  float* AcOut    = fout + NN;                        // [NN, NN]
  float* mask_out = AcOut + (size_t)NN * NN;          // [NN]

  char* w = (char*)d_ws;
  auto alloc = [&](size_t bytes) -> char* {
    char* p = w; w += (bytes + 255) & ~(size_t)255; return p;
  };
  // --- zero-init block (contiguous; one fill) ---
  char* zstart   = w;
  float* Ad      = (float*)alloc((size_t)NN * NN * 4);
  float* Sd      = (float*)alloc((size_t)NN * NN * 4);
  float* xc      = (float*)alloc((size_t)NN * C2 * 4);
  float* ssum    = (float*)alloc(NN * 4);
  float* fitacc  = (float*)alloc(NN * LEO * 4);
  int*   indeg   = (int*)  alloc(NN * 4);
  int*   wmbuf   = (int*)  alloc(NN * 4);
  int*   flags   = (int*)  alloc(256);
  size_t zbytes  = (size_t)(w - zstart);
  // --- -inf init block (contiguous; one fill) ---
  char* nstart   = w;
  float* xq_t    = (float*)alloc((size_t)NN * CD * 4);
  float* smax    = (float*)alloc(NN * 4);
  size_t nbytes  = (size_t)(w - nstart);
  // --- no-init block (fully overwritten each call) ---
  float* T1       = (float*)alloc((size_t)NN * NN * 4);
  float* xq_s     = (float*)alloc((size_t)NN * CD * 4);
  float* xqlin_t  = (float*)alloc((size_t)NN * CD * 4);
  float* xqlin_s  = (float*)alloc((size_t)NN * CD * 4);
  float* scores   = (float*)alloc(EE * 4);
  float* aq_t     = (float*)alloc(NN * 4);
  float* aj_t     = (float*)alloc(NN * 4);
  float* aq_s     = (float*)alloc(NN * 4);
  float* aj_s     = (float*)alloc(NN * 4);
  float* h1_t     = (float*)alloc(NN * LEO * 4);
  float* h1_s     = (float*)alloc(NN * LEO * 4);
  float* fitbase_t= (float*)alloc(NN * LEO * 4);
  float* fitbase_s= (float*)alloc(NN * LEO * 4);
  float* fit      = (float*)alloc(NN * FD * 4);
  float* fr       = (float*)alloc(NN * 4);
  float* gate     = (float*)alloc(256);
  float* maskf    = (float*)alloc(NN * 4);
  int*   maski    = (int*)  alloc(NN * 4);
  (void)ws_size; (void)n_in; (void)in_sizes; (void)out_size;

  const int T = 256;
  {
    int zn = (int)(zbytes / 4);
    fill_f32<<<(zn + T - 1) / T, T, 0, stream>>>((float*)zstart, 0.0f, zn);
    int nn = (int)(nbytes / 4);
    fill_f32<<<(nn + T - 1) / T, T, 0, stream>>>((float*)nstart, -INFINITY, nn);
  }
  // segment max features
  chain_xq_kernel<<<(NN * CD) / T, T, 0, stream>>>(x, xq_s);
  scatter_max_kernel<<<(EE * CD) / T, T, 0, stream>>>(ei, x, xq_t);
  fix_neginf<<<(NN * CD) / T, T, 0, stream>>>(xq_t, NN * CD);
  // xqlin = xq @ lin_w + lin_b  (M=2048, K=256, N=256)
  {
    dim3 grid(CD / BN, NN / BM);
    wmma_gemm_f32<<<grid, 256, 0, stream>>>(xq_t, CD, 0, lin_t_w, CD, CD,
                                            lin_t_b, nullptr, nullptr, 0, xqlin_t, CD);
    wmma_gemm_f32<<<grid, 256, 0, stream>>>(xq_s, CD, 0, lin_s_w, CD, CD,
                                            lin_s_b, nullptr, nullptr, 0, xqlin_s, CD);
  }
  // attention scalars and softmax (random graph)
  node_dots_kernel<<<(NN * 4) / 128, 128, 0, stream>>>(x, xqlin_t, xqlin_s,
                                                       att_t_w, att_s_w,
                                                       aq_t, aj_t, aq_s, aj_s);
  edge_scoreA<<<EE / T, T, 0, stream>>>(ei, aq_t, aj_t, att_t_b, scores, smax, indeg);
  fix_neginf<<<(NN + T - 1) / T, T, 0, stream>>>(smax, NN);
  edge_scoreB<<<EE / T, T, 0, stream>>>(ei, scores, smax, ssum);
  edge_scatter_xt<<<EE, 256, 0, stream>>>(ei, x, scores, ssum, xc);
  // chain attention + aggregation
  chain_xs<<<NN, 256, 0, stream>>>(x, aq_s, aj_s, att_s_b, xc);
  // LEConv
  node_h_kernel<<<NN, 64, 0, stream>>>(xc, le_t_w1, le_t_b1, le_t_w2, le_t_w3, le_t_b3,
                                       le_s_w1, le_s_b1, le_s_w2, le_s_w3, le_s_b3,
                                       indeg, h1_t, h1_s, fitbase_t, fitbase_s);
  edge_scatter_fit<<<(EE * LEO) / T, T, 0, stream>>>(ei, h1_t, fitacc);
  fit_assemble<<<(NN * FD) / T, T, 0, stream>>>(fitacc, fitbase_t, fitbase_s, h1_s, fit);
  // SE gate, f, conv
  se_gate_kernel<<<1, 256, 0, stream>>>(fit, se_w1, se_b1, se_w2, se_b2, gate);
  fr_kernel<<<NN / T, T, 0, stream>>>(fit, gate, fr);
  conv_kernel<<<NN / T, T, 0, stream>>>(fr, conv_w, conv_b, fout);
  // peak / window-max / mask
  window_scatter<<<NN / T, T, 0, stream>>>(fout, wmbuf);
  argmax_kernel<<<1, 256, 0, stream>>>(fout, flags);
  mask_combine<<<NN / T, T, 0, stream>>>(fout, wmbuf, maski, flags);
  mask_final<<<NN / T, T, 0, stream>>>(maski, flags, maskf, mask_out);
  // x_out
  xout_kernel<<<(NN * C2) / T, T, 0, stream>>>(xc, fout, maskf, xout);
  // dense A, S and coarsened adjacency: Ac = (S*mask_col)^T A (S*mask_col), diag zeroed
  build_AS<<<EE / T, T, 0, stream>>>(ei, ew, scores, Ad, Sd);
  {
    dim3 grid(NN / BN, NN / BM);
    // T1 = S^T @ A (mask folded into final epilogue)
    wmma_gemm_f32<<<grid, 256, 0, stream>>>(Sd, NN, 1, Ad, NN, NN,
                                            nullptr, nullptr, nullptr, 0, T1, NN);
    // Ac = mask_row * mask_col * (T1 @ S), diag -> 0
    wmma_gemm_f32<<<grid, 256, 0, stream>>>(T1, NN, 0, Sd, NN, NN,
                                            nullptr, maskf, maskf, 1, AcOut, NN);
  }
}